// SpectralNetLoss_86758339379794
// MI455X (gfx1250) — compile-verified
//
#include <hip/hip_runtime.h>
#include <hip/hip_bf16.h>

typedef float v2f __attribute__((ext_vector_type(2)));
typedef float v8f __attribute__((ext_vector_type(8)));

#define M_DIM 8192
#define K_DIM 32

__global__ void __launch_bounds__(64)
zero_out_kernel(float* out, int n) {
    int i = blockIdx.x * blockDim.x + threadIdx.x;
    if (i < n) out[i] = 0.0f;
}

// One block = one 16-row strip of W. 8 waves split the K=8192 contraction.
// Per wave k-step (4 columns of W):
//   A frag (16x4 fp32): lane<16 -> rows r0..r0+15, K={j,j+1}; lane>=16 -> K={j+2,j+3}
//   B frags (4x16 fp32) from Y, two N-tiles (cols 0-15, 16-31)
//   2x v_wmma_f32_16x16x4_f32 accumulate the 16x32 WY tile in fp32
//   row sums of W accumulate on the same loaded data (VALU, free vs. memory)
__global__ void __launch_bounds__(256)
spectral_loss_wmma_kernel(const float* __restrict__ W,
                          const float* __restrict__ Y,
                          float* __restrict__ out) {
    const int tid  = threadIdx.x;
    const int lane = tid & 31;
    const int wv   = tid >> 5;        // wave 0..7
    const int rl   = lane & 15;       // row-in-strip (A) / col-in-tile (B)
    const int h    = lane >> 4;       // lane-half -> K pair select
    const int r0   = blockIdx.x * 16; // strip base row

    v8f acc0 = {0.f,0.f,0.f,0.f,0.f,0.f,0.f,0.f}; // WY cols 0-15
    v8f acc1 = {0.f,0.f,0.f,0.f,0.f,0.f,0.f,0.f}; // WY cols 16-31
    float accD0 = 0.0f, accD1 = 0.0f;              // split row-sum chains

    const float* wrow = W + (size_t)(r0 + rl) * M_DIM;
    const int    kb   = wv * 4 + 2 * h;            // this lane's base column

    // 256 k-steps per wave; guaranteed 4x unroll via constant inner loop.
    for (int t = 0; t < 64; ++t) {
#pragma unroll
        for (int u = 0; u < 4; ++u) {
            const int ja = kb + ((t * 4 + u) << 5);   // +32 cols per step
            float2 wa = *(const float2*)(wrow + ja);  // global_load_b64
            v2f a = {wa.x, wa.y};
            if (u & 1) accD1 += wa.x + wa.y;
            else       accD0 += wa.x + wa.y;

            const float* yb = Y + (size_t)ja * K_DIM + rl;
            v2f b0 = {yb[0],  yb[K_DIM]};             // Y[ja][c], Y[ja+1][c]
            v2f b1 = {yb[16], yb[16 + K_DIM]};        // c = 16+rl

            if (u == 0) {                             // folds at compile time
                int jp = ja + 128;                    // 4 wave-steps ahead
                jp = (jp < M_DIM) ? jp : ja;          // v_cndmask, no branch
                __builtin_prefetch(wrow + jp, 0, 3);  // global_prefetch_b8
            }

            acc0 = __builtin_amdgcn_wmma_f32_16x16x4_f32(
                       false, a, false, b0, (short)0, acc0, false, false);
            acc1 = __builtin_amdgcn_wmma_f32_16x16x4_f32(
                       false, a, false, b1, (short)0, acc1, false, false);
        }
    }
    const float accD = accD0 + accD1;

    // ---- cross-wave reduction through LDS ----
    __shared__ float lds_acc[8][16][32];  // [wave][accslot (2 tiles x 8 vgpr)][lane]
    __shared__ float lds_d[8][32];        // [wave][lane] partial row sums
    __shared__ float lds_r[8];            // final scalar reduce
#pragma unroll
    for (int v = 0; v < 8; ++v) {
        lds_acc[wv][v][lane]     = acc0[v];
        lds_acc[wv][8 + v][lane] = acc1[v];
    }
    lds_d[wv][lane] = accD;
    __syncthreads();

    // 512 WY elements in the strip (16 rows x 32 cols); each thread folds 2.
    // C/D layout: VGPR v, lane<16 -> (row=v,    col=lane)
    //                    lane>=16 -> (row=v+8,  col=lane-16)
    float part = 0.0f;
#pragma unroll
    for (int it = 0; it < 2; ++it) {
        const int s  = tid + it * 256;    // 0..511
        const int v  = s >> 5;            // acc slot 0..15
        const int l  = s & 31;            // lane slot
        float wy = 0.0f;
#pragma unroll
        for (int u = 0; u < 8; ++u) wy += lds_acc[u][v][l];

        const int tile = v >> 3;
        const int vg   = v & 7;
        const int r    = vg + ((l & 16) ? 8 : 0);
        const int c    = tile * 16 + (l & 15);

        float dr = 0.0f;
#pragma unroll
        for (int u = 0; u < 8; ++u) dr += lds_d[u][r] + lds_d[u][r + 16];

        const float y = Y[(size_t)(r0 + r) * K_DIM + c];
        part += y * (y - wy / dr);        // y*LY fused epilogue
    }

    // block-reduce 'part' (wave32 shuffles, then one LDS hop)
#pragma unroll
    for (int off = 16; off > 0; off >>= 1)
        part += __shfl_down(part, off, 32);
    if (lane == 0) lds_r[wv] = part;
    __syncthreads();
    if (tid == 0) {
        float tot = 0.0f;
#pragma unroll
        for (int u = 0; u < 8; ++u) tot += lds_r[u];
        atomicAdd(out, tot * (1.0f / (float)M_DIM));
    }
}

extern "C" void kernel_launch(void* const* d_in, const int* in_sizes, int n_in,
                              void* d_out, int out_size, void* d_ws, size_t ws_size,
                              hipStream_t stream) {
    const float* W = (const float*)d_in[0];
    const float* Y = (const float*)d_in[1];
    float* out = (float*)d_out;

    zero_out_kernel<<<1, 64, 0, stream>>>(out, out_size);

    dim3 grid(M_DIM / 16);   // 512 strips
    dim3 block(256);         // 8 waves
    spectral_loss_wmma_kernel<<<grid, block, 0, stream>>>(W, Y, out);
}